// SobelLoss_43104291783230
// MI455X (gfx1250) — compile-verified
//
#include <hip/hip_runtime.h>

// Problem constants from the reference: (B, C, H, W) = (32, 1, 1024, 1024), fp32.
#define BATCH   32
#define HEIGHT  1024
#define WIDTH   1024
#define SEGS    8                       // row segments per image column
#define SEGROWS (HEIGHT / SEGS)         // 128 rows per thread
#define TPB     256
#define NBLOCKS (BATCH * SEGS * WIDTH / TPB)   // 1024 blocks, exact fit -> EXEC all-1s

typedef float v2f __attribute__((ext_vector_type(2)));
typedef float v8f __attribute__((ext_vector_type(8)));

// Broadcast value held in `srcLane` to all lanes (wave32) via LDS-permute HW.
__device__ __forceinline__ float lane_bcast_f32(float v, int srcLane) {
    int r = __builtin_amdgcn_ds_bpermute(srcLane << 2, __float_as_int(v));
    return __int_as_float(r);
}

// Full wave32 sum using V_WMMA_F32_16X16X4_F32 with B == ones.
// A (16x4 fp32) holds one value per lane in VGPR0 (A[m][0] = lane m,
// A[m][2] = lane m+16), zeros in VGPR1. D[m][n] = sum_k A[m][k] = v_m + v_{m+16}.
// Per-lane sum of the 8 D VGPRs yields rows 0-7 in lanes 0-15 and rows 8-15 in
// lanes 16-31; total = lane0 + lane16.
__device__ __forceinline__ float wave_sum_wmma(float v) {
    v2f a; a[0] = v;    a[1] = 0.0f;
    v2f b; b[0] = 1.0f; b[1] = 1.0f;   // all-ones B: result independent of B layout
    v8f c = {};
    v8f d = __builtin_amdgcn_wmma_f32_16x16x4_f32(
        /*neg_a=*/false, a, /*neg_b=*/false, b,
        /*c_mod=*/(short)0, c, /*reuse_a=*/false, /*reuse_b=*/false);
    float s = d[0] + d[1] + d[2] + d[3] + d[4] + d[5] + d[6] + d[7];
    return lane_bcast_f32(s, 0) + lane_bcast_f32(s, 16);
}

__global__ void sobel_loss_main(const float* __restrict__ fake,
                                const float* __restrict__ real,
                                float2* __restrict__ block_part) {
    const int t   = blockIdx.x * TPB + threadIdx.x;     // exact-fit grid, no guard
    const int w   = t % WIDTH;                          // column (lane-contiguous)
    const int seg = (t / WIDTH) % SEGS;
    const int b   = t / (WIDTH * SEGS);

    const float* __restrict__ fimg = fake + (size_t)b * HEIGHT * WIDTH;
    const float* __restrict__ rimg = real + (size_t)b * HEIGHT * WIDTH;

    // symmetric padding == clamped neighbor columns
    const int wm = (w > 0) ? (w - 1) : 0;
    const int wp = (w < WIDTH - 1) ? (w + 1) : (WIDTH - 1);

    const int h0 = seg * SEGROWS;
    const int rt = (h0 > 0) ? (h0 - 1) : 0;             // clamped top halo row

    // 3-row register pipeline for columns wm and wp of both images
    float fa_m = fimg[(size_t)rt * WIDTH + wm], fa_p = fimg[(size_t)rt * WIDTH + wp];
    float ra_m = rimg[(size_t)rt * WIDTH + wm], ra_p = rimg[(size_t)rt * WIDTH + wp];
    float fb_m = fimg[(size_t)h0 * WIDTH + wm], fb_p = fimg[(size_t)h0 * WIDTH + wp];
    float rb_m = rimg[(size_t)h0 * WIDTH + wm], rb_p = rimg[(size_t)h0 * WIDTH + wp];

    float num = 0.0f, cnt = 0.0f;

#pragma unroll 4
    for (int h = h0; h < h0 + SEGROWS; ++h) {
        const int rp = (h < HEIGHT - 1) ? (h + 1) : (HEIGHT - 1);  // clamped bottom row
        const size_t ro = (size_t)rp * WIDTH;
        const float fd_m = fimg[ro + wm], fd_p = fimg[ro + wp];
        const float rd_m = rimg[ro + wm], rd_p = rimg[ro + wp];

        const float fs = (fa_p + 2.0f * fb_p + fd_p) - (fa_m + 2.0f * fb_m + fd_m);
        const float rs = (ra_p + 2.0f * rb_p + rd_p) - (ra_m + 2.0f * rb_m + rd_m);

        if (rs > 0.0f) { num += fabsf(fs - rs); cnt += 1.0f; }

        fa_m = fb_m; fa_p = fb_p; fb_m = fd_m; fb_p = fd_p;
        ra_m = rb_m; ra_p = rb_p; rb_m = rd_m; rb_p = rd_p;
    }

    // wave32 reduction via WMMA (EXEC is all-1s: exact-fit grid, no early exits)
    const float wnum = wave_sum_wmma(num);
    const float wcnt = wave_sum_wmma(cnt);

    __shared__ float snum[TPB / 32];
    __shared__ float scnt[TPB / 32];
    const int wave = threadIdx.x >> 5;
    const int lane = threadIdx.x & 31;
    if (lane == 0) { snum[wave] = wnum; scnt[wave] = wcnt; }
    __syncthreads();

    if (threadIdx.x == 0) {
        float n = 0.0f, c = 0.0f;
        for (int i = 0; i < TPB / 32; ++i) { n += snum[i]; c += scnt[i]; }
        block_part[blockIdx.x] = make_float2(n, c);
    }
}

__global__ void sobel_loss_finalize(const float2* __restrict__ part,
                                    float* __restrict__ out) {
    __shared__ float2 acc[TPB];
    float n = 0.0f, c = 0.0f;
    for (int i = threadIdx.x; i < NBLOCKS; i += TPB) {
        float2 p = part[i];
        n += p.x; c += p.y;
    }
    acc[threadIdx.x] = make_float2(n, c);
    __syncthreads();
    if (threadIdx.x == 0) {
        float N = 0.0f, C = 0.0f;
        for (int i = 0; i < TPB; ++i) { N += acc[i].x; C += acc[i].y; }
        out[0] = N / C / (float)BATCH;
    }
}

extern "C" void kernel_launch(void* const* d_in, const int* in_sizes, int n_in,
                              void* d_out, int out_size, void* d_ws, size_t ws_size,
                              hipStream_t stream) {
    (void)in_sizes; (void)n_in; (void)out_size; (void)ws_size;
    const float* fake = (const float*)d_in[0];
    const float* real = (const float*)d_in[1];
    float2* parts = (float2*)d_ws;            // NBLOCKS * sizeof(float2) = 8 KB
    float*  out   = (float*)d_out;

    sobel_loss_main<<<dim3(NBLOCKS), dim3(TPB), 0, stream>>>(fake, real, parts);
    sobel_loss_finalize<<<dim3(1), dim3(TPB), 0, stream>>>(parts, out);
}